// Renderer_68496138436786
// MI455X (gfx1250) — compile-verified
//
#include <hip/hip_runtime.h>
#include <hip/hip_bf16.h>
#include <stdint.h>

// Point-splat depth render (z-buffer min-scatter) for MI455X / gfx1250.
//
// Strategy: scatter with device-scope FP atomic-min.
//  - Output (19.66 MB) fits in the 192 MB L2, so every atomic is an L2-side
//    RMW; average contention is ~1 atomic per pixel. Traffic floor is the
//    one-time 19.66 MB output write (~0.84 us at 23.3 TB/s).
//  - Uses the CDNA5/GFX12 instruction GLOBAL_ATOMIC_MIN_NUM_F32 (IEEE
//    minimumNumber, no CAS loop) via inline asm with scope:SCOPE_DEV
//    (validated in round 1: assembles and appears in the vmem histogram).
//  - Dead-candidate elimination: with the window anchored at floor(px),
//    the dx=-3 column, dy=-3 row, and the four {-2,3}x{-2,3} corners can
//    never satisfy fx^2+fy^2 <= 6.25 (min distance^2 is 9, 9, and 8).
//    Only 32 of the reference's 49 candidates are live; the float math for
//    those is identical to the reference, so the output is bit-exact.
//  - No matmul structure exists in this problem, so WMMA is intentionally
//    not used; the FP min atomic is the CDNA5-specific path that matters.

#define IMG_H 480
#define IMG_W 640
#define RADIUS2 6.25f   // 2.5^2
#define BG 1.0f

// Device-scope, non-returning FP32 atomic min at L2 (CDNA5 naming: *_min_num_*).
// Non-returning => tracked on STOREcnt; s_endpgm's implicit wait-idle covers it.
__device__ __forceinline__ void global_atomic_min_f32(float* p, float v) {
    asm volatile("global_atomic_min_num_f32 %0, %1, off scope:SCOPE_DEV"
                 :
                 : "v"((unsigned long long)(uintptr_t)p), "v"(v)
                 : "memory");
}

// ---------------------------------------------------------------------------
// Kernel 1: fill z-buffer with BG using 128-bit stores (hits the write-
// bandwidth floor; default RT temporal hint keeps lines in L2 so the
// subsequent atomics hit).
// ---------------------------------------------------------------------------
__global__ void zbuf_init_kernel(float* __restrict__ out, int n) {
    int i = (blockIdx.x * blockDim.x + threadIdx.x) * 4;
    if (i + 3 < n) {
        float4 bg = make_float4(BG, BG, BG, BG);
        *reinterpret_cast<float4*>(out + i) = bg;   // global_store_b128
    } else {
        for (int k = i; k < n; ++k) out[k] = BG;    // tail (out_size % 4 == 0)
    }
}

// ---------------------------------------------------------------------------
// Kernel 2: one thread per point. Unrolled 6x6 live window minus 4 dead
// corners (32 candidates), identical float math to the reference
// (fx*fx + fy*fy <= 6.25f with fx = (float)ix - px), bounds-checked,
// then FP atomic min.
// ---------------------------------------------------------------------------
__global__ void splat_kernel(const float* __restrict__ pts,  // (B*N, 3) flat
                             float* __restrict__ out,        // (B, H, W) flat
                             int npts, int n_per_batch) {
    int tid = blockIdx.x * blockDim.x + threadIdx.x;
    if (tid >= npts) return;

    // 12-byte struct load (compiler coalesces to b96 / b64+b32).
    float px = pts[3 * tid + 0];
    float py = pts[3 * tid + 1];
    float pz = pts[3 * tid + 2];

    int b = tid / n_per_batch;                     // batch index (once/thread)
    float* img = out + (size_t)b * (IMG_H * IMG_W);

    int bx = (int)floorf(px);
    int by = (int)floorf(py);

#pragma unroll
    for (int dy = -2; dy <= 3; ++dy) {
        int iy = by + dy;
        if ((unsigned)iy >= (unsigned)IMG_H) continue;
        float fy  = (float)iy - py;
        float fy2 = fy * fy;
        float* row = img + iy * IMG_W;
#pragma unroll
        for (int dx = -2; dx <= 3; ++dx) {
            // Compile-time skip of the 4 provably-dead corners.
            if ((dx == -2 || dx == 3) && (dy == -2 || dy == 3)) continue;
            int ix = bx + dx;
            if ((unsigned)ix < (unsigned)IMG_W) {
                float fx = (float)ix - px;
                if (fx * fx + fy2 <= RADIUS2) {
                    global_atomic_min_f32(row + ix, pz);  // EXEC-masked atomic
                }
            }
        }
    }
}

// ---------------------------------------------------------------------------
// Host entry. Inputs: d_in[0] = points (B, N, 3) f32. Output: (B, H, W) f32.
// ---------------------------------------------------------------------------
extern "C" void kernel_launch(void* const* d_in, const int* in_sizes, int n_in,
                              void* d_out, int out_size, void* d_ws, size_t ws_size,
                              hipStream_t stream) {
    const float* pts = (const float*)d_in[0];
    float* out = (float*)d_out;

    const int npts = in_sizes[0] / 3;                       // B * N
    const int batches = out_size / (IMG_H * IMG_W);         // B
    const int n_per_batch = (batches > 0) ? (npts / batches) : npts;  // N

    // 1) z-buffer = BG
    {
        int n_vec_threads = (out_size + 3) / 4;
        int blocks = (n_vec_threads + 255) / 256;
        zbuf_init_kernel<<<blocks, 256, 0, stream>>>(out, out_size);
    }

    // 2) scatter-min splat
    {
        int blocks = (npts + 255) / 256;
        splat_kernel<<<blocks, 256, 0, stream>>>(pts, out, npts, n_per_batch);
    }
}